// SpGraph_Mul_AttentionLayer_13374528160098
// MI455X (gfx1250) — compile-verified
//
#include <hip/hip_runtime.h>
#include <hip/hip_bf16.h>

#define N_NODES 50000
#define E_EDGES 1600000
#define F_IN    128
#define F_OUT   64
#define ALPHA   0.2f

typedef __attribute__((ext_vector_type(2))) float v2f;
typedef __attribute__((ext_vector_type(8))) float v8f;

// ---------------------------------------------------------------------------
// GEMM via V_WMMA_F32_16X16X4_F32 (full fp32 precision).
// One wave computes one 16(M)x16(N) tile; 4 waves/block cover all 64 cols.
// grid.x = 3125 row tiles (50000/16 exact), grid.y selects input matrix.
// ---------------------------------------------------------------------------
__global__ __launch_bounds__(128) void gat_gemm_wmma(
    const float* __restrict__ X0, const float* __restrict__ X1,
    const float* __restrict__ X2, const float* __restrict__ W,
    float* __restrict__ Y0, float* __restrict__ Y1, float* __restrict__ Y2)
{
    const float* X; float* Y;
    if      (blockIdx.y == 0) { X = X0; Y = Y0; }
    else if (blockIdx.y == 1) { X = X1; Y = Y1; }
    else                      { X = X2; Y = Y2; }

    const int wave  = threadIdx.x >> 5;
    const int lane  = threadIdx.x & 31;
    const int laneM = lane & 15;        // M for A-frag / N for B-frag / N for C
    const int half  = lane >> 4;        // selects K pair {0,1} vs {2,3}
    const int m0    = blockIdx.x * 16;
    const int n0    = wave * 16;

    // A 16x4 layout: v0 = K=k0+2*half, v1 = K=k0+2*half+1  (contiguous pair)
    const float* arow = X + (size_t)(m0 + laneM) * F_IN + 2 * half;
    // B 4x16 layout: v0 = row K=k0+2*half, v1 = row K=k0+2*half+1, col = laneM
    const float* bcol = W + (size_t)(2 * half) * F_OUT + n0 + laneM;

    v8f acc = {};
#pragma unroll 8
    for (int k0 = 0; k0 < F_IN; k0 += 4) {
        v2f a, b;
        a.x = arow[k0];
        a.y = arow[k0 + 1];
        b.x = bcol[(size_t)k0 * F_OUT];
        b.y = bcol[(size_t)(k0 + 1) * F_OUT];
        acc = __builtin_amdgcn_wmma_f32_16x16x4_f32(
            false, a, false, b, (short)0, acc, false, false);
    }

    // C/D layout: VGPR r -> M = r + 8*half, N = laneM
    float* yout = Y + (size_t)(m0 + 8 * half) * F_OUT + n0 + laneM;
#pragma unroll
    for (int r = 0; r < 8; ++r) yout[(size_t)r * F_OUT] = acc[r];
}

// ---------------------------------------------------------------------------
__device__ __forceinline__ float wave_reduce_add(float v) {
#pragma unroll
    for (int off = 16; off > 0; off >>= 1) v += __shfl_xor(v, off, 32);
    return v;   // all 32 lanes hold the sum (butterfly)
}

__device__ __forceinline__ float leaky_neg_exp(float s) {
    float lr = (s >= 0.f) ? s : ALPHA * s;
    return expf(-lr);
}

// ---------------------------------------------------------------------------
// Zero the atomic-accumulation region (hp_acc + segmax + segsum + rowsum).
// ---------------------------------------------------------------------------
__global__ void zero_kernel(float* __restrict__ p, long long n) {
    long long i = (long long)blockIdx.x * blockDim.x + threadIdx.x;
    long long stride = (long long)gridDim.x * blockDim.x;
    for (; i < n; i += stride) p[i] = 0.f;
}

// ---------------------------------------------------------------------------
// Per-node column score: cs[i] = exp(-leaky(h[ec0[i]]·a1l + ph[i]·a1r))
// plus segment-max via int atomicMax (values are > 0, init = 0 bits).
// One wave per node: coalesced float2 row loads.
// ---------------------------------------------------------------------------
__global__ __launch_bounds__(256) void col_score_kernel(
    const float* __restrict__ h,  const float* __restrict__ ph,
    const int* __restrict__ ec0,  const int* __restrict__ row_i,
    const float* __restrict__ a1,
    float* __restrict__ cs, float* __restrict__ segmax)
{
    const int i = blockIdx.x * 8 + (threadIdx.x >> 5);
    if (i >= N_NODES) return;
    const int lane = threadIdx.x & 31;
    const int f    = lane * 2;
    const int src  = ec0[i];
    float2 hv = *(const float2*)(h  + (size_t)src * F_OUT + f);
    float2 pv = *(const float2*)(ph + (size_t)i   * F_OUT + f);
    float partial = hv.x * a1[f] + hv.y * a1[f + 1]
                  + pv.x * a1[F_OUT + f] + pv.y * a1[F_OUT + f + 1];
    float s = wave_reduce_add(partial);
    if (lane == 0) {
        float v = leaky_neg_exp(s);
        cs[i] = v;
        atomicMax((int*)&segmax[row_i[i]], __float_as_int(v));
    }
}

__global__ void col_softmax_sum(
    const float* __restrict__ cs, const int* __restrict__ row_i,
    const float* __restrict__ segmax,
    float* __restrict__ exarr, float* __restrict__ segsum)
{
    int i = blockIdx.x * blockDim.x + threadIdx.x;
    if (i >= N_NODES) return;
    float ex = expf(cs[i] - segmax[row_i[i]]);
    exarr[i] = ex;
    atomicAdd(&segsum[row_i[i]], ex);
}

__global__ void col_softmax_norm(
    const float* __restrict__ exarr, const int* __restrict__ row_i,
    const float* __restrict__ segsum, float* __restrict__ ecol)
{
    int i = blockIdx.x * blockDim.x + threadIdx.x;
    if (i >= N_NODES) return;
    ecol[i] = exarr[i] / (segsum[row_i[i]] + 1e-16f);
}

// ---------------------------------------------------------------------------
// Per-edge pass: one wave32 per edge. Coalesced 256B row gathers, butterfly
// dot-reduction, atomic rowsum + atomic weighted scatter into hp_acc.
// ---------------------------------------------------------------------------
__global__ __launch_bounds__(256) void edge_kernel(
    const float* __restrict__ h,  const float* __restrict__ nh,
    const int* __restrict__ e0,   const int* __restrict__ e1,
    const int* __restrict__ rr,   const float* __restrict__ a2,
    const float* __restrict__ ecol,
    float* __restrict__ edge_e, float* __restrict__ rowsum,
    float* __restrict__ hp_acc)
{
    const long long e = (long long)blockIdx.x * 8 + (threadIdx.x >> 5);
    if (e >= E_EDGES) return;
    const int lane = threadIdx.x & 31;
    const int f    = lane * 2;
    const int dst  = e0[e];
    const int src  = e1[e];
    const int r    = rr[e];
    float2 hv = *(const float2*)(h  + (size_t)src * F_OUT + f);
    float2 nv = *(const float2*)(nh + (size_t)r   * F_OUT + f);
    float partial = nv.x * a2[f] + nv.y * a2[f + 1]
                  + hv.x * a2[F_OUT + f] + hv.y * a2[F_OUT + f + 1];
    float s  = wave_reduce_add(partial);
    float ee = leaky_neg_exp(s) * ecol[r];
    if (lane == 0) {
        edge_e[e] = ee;
        atomicAdd(&rowsum[dst], ee);
    }
    atomicAdd(&hp_acc[(size_t)dst * F_OUT + f],     ee * hv.x);
    atomicAdd(&hp_acc[(size_t)dst * F_OUT + f + 1], ee * hv.y);
}

// ---------------------------------------------------------------------------
// Finalize: h_prime = elu(hp_acc / rowsum'), rowsum' = (rowsum==0 ? 1 : rowsum)
// ---------------------------------------------------------------------------
__global__ void finalize_nodes(
    const float* __restrict__ hp_acc, const float* __restrict__ rowsum,
    float* __restrict__ rs_final, float* __restrict__ out_h)
{
    int idx = blockIdx.x * blockDim.x + threadIdx.x;   // over N*F_OUT
    if (idx >= N_NODES * F_OUT) return;
    int n = idx >> 6;
    float s = rowsum[n];
    float d = (s == 0.f) ? 1.f : s;
    if ((idx & 63) == 0) rs_final[n] = d;
    float v = hp_acc[idx] / d;
    out_h[idx] = (v > 0.f) ? v : (expf(v) - 1.f);
}

__global__ void finalize_edges(
    const float* __restrict__ edge_e, const int* __restrict__ e0,
    const float* __restrict__ rs_final, const int* __restrict__ edge_all,
    float* __restrict__ out_edge, float* __restrict__ out_att)
{
    long long e = (long long)blockIdx.x * blockDim.x + threadIdx.x;
    if (e >= 2LL * E_EDGES) return;
    out_edge[e] = (float)edge_all[e];
    if (e < E_EDGES) out_att[e] = edge_e[e] / rs_final[e0[e]];
}

// ---------------------------------------------------------------------------
extern "C" void kernel_launch(void* const* d_in, const int* in_sizes, int n_in,
                              void* d_out, int out_size, void* d_ws, size_t ws_size,
                              hipStream_t stream) {
    const float* input      = (const float*)d_in[0];
    const float* p_h        = (const float*)d_in[1];
    const float* new_h      = (const float*)d_in[2];
    const int*   edge       = (const int*)d_in[3];   // (2, E)
    const int*   edge_col   = (const int*)d_in[4];   // (2, N)
    const int*   row_i      = (const int*)d_in[5];   // (N,)
    const int*   row_resort = (const int*)d_in[6];   // (E,)
    const float* W          = (const float*)d_in[8]; // (128, 64)
    const float* a1         = (const float*)d_in[9]; // (1, 128)
    const float* a2         = (const float*)d_in[10];// (1, 128)

    const int* e0  = edge;
    const int* e1  = edge + E_EDGES;
    const int* ec0 = edge_col;

    // Workspace layout (floats). Atomic region first so one zero pass covers it.
    float* ws      = (float*)d_ws;
    float* hp_acc  = ws;                       // N*64
    float* segmax  = hp_acc + (size_t)N_NODES * F_OUT;  // N
    float* segsum  = segmax + N_NODES;                  // N
    float* rowsum  = segsum + N_NODES;                  // N
    float* h       = rowsum + N_NODES;                  // N*64
    float* ph      = h  + (size_t)N_NODES * F_OUT;      // N*64
    float* nh      = ph + (size_t)N_NODES * F_OUT;      // N*64
    float* cs      = nh + (size_t)N_NODES * F_OUT;      // N
    float* exarr   = cs + N_NODES;                      // N
    float* ecol    = exarr + N_NODES;                   // N
    float* rs_fin  = ecol + N_NODES;                    // N
    float* edge_e  = rs_fin + N_NODES;                  // E

    float* out_h    = (float*)d_out;
    float* out_edge = out_h + (size_t)N_NODES * F_OUT;
    float* out_att  = out_edge + 2LL * E_EDGES;

    // 1. zero atomic accumulators (N*64 + 3N contiguous floats)
    zero_kernel<<<4096, 256, 0, stream>>>(hp_acc, (long long)N_NODES * (F_OUT + 3));

    // 2. three GEMMs via fp32 WMMA: 3125 row tiles x 3 matrices
    gat_gemm_wmma<<<dim3(N_NODES / 16, 3), 128, 0, stream>>>(
        input, p_h, new_h, W, h, ph, nh);

    // 3. column scores + segment max (wave per node)
    col_score_kernel<<<(N_NODES + 7) / 8, 256, 0, stream>>>(
        h, ph, ec0, row_i, a1, cs, segmax);

    // 4-5. segment softmax
    col_softmax_sum<<<(N_NODES + 255) / 256, 256, 0, stream>>>(
        cs, row_i, segmax, exarr, segsum);
    col_softmax_norm<<<(N_NODES + 255) / 256, 256, 0, stream>>>(
        exarr, row_i, segsum, ecol);

    // 6. edge pass (wave per edge)
    edge_kernel<<<(E_EDGES + 7) / 8, 256, 0, stream>>>(
        h, nh, e0, e1, row_resort, a2, ecol, edge_e, rowsum, hp_acc);

    // 7. finalize nodes: divide + ELU
    finalize_nodes<<<(N_NODES * F_OUT + 255) / 256, 256, 0, stream>>>(
        hp_acc, rowsum, rs_fin, out_h);

    // 8. finalize edges: attention + edge copy
    finalize_edges<<<(2 * E_EDGES + 255) / 256, 256, 0, stream>>>(
        edge_e, e0, rs_fin, edge, out_edge, out_att);
}